// EpistemicQuantizer_17875653886595
// MI455X (gfx1250) — compile-verified
//
#include <hip/hip_runtime.h>
#include <hip/hip_bf16.h>

typedef float v2f __attribute__((ext_vector_type(2)));
typedef float v8f __attribute__((ext_vector_type(8)));

#define DTOK 64
#define NEG_INF (-3.402823466e38f)

// ---------------------------------------------------------------------------
// Kernel 1: reciprocal L2 norms per row (row length = 64 floats).
// rn[r] = 1 / max(||src[r,:]||_2, 1e-12)   (matches F.normalize eps semantics)
// ---------------------------------------------------------------------------
__global__ void eq_rnorm_kernel(const float* __restrict__ src,
                                float* __restrict__ rn, int rows) {
    int r = blockIdx.x * blockDim.x + threadIdx.x;
    if (r >= rows) return;
    const float4* p = (const float4*)(src + (size_t)r * DTOK);
    float ss = 0.f;
#pragma unroll
    for (int i = 0; i < DTOK / 4; ++i) {
        float4 v = p[i];
        ss += v.x * v.x + v.y * v.y + v.z * v.z + v.w * v.w;
    }
    float nrm = fmaxf(sqrtf(ss), 1e-12f);
    rn[r] = 1.0f / nrm;
}

// ---------------------------------------------------------------------------
// Kernel 2: fused cosine-sim GEMM (WMMA f32 16x16x4) + argmax + gather.
// One wave32 per 16-row tile of x. Loops over V in 16-column chunks.
// ---------------------------------------------------------------------------
__global__ __launch_bounds__(32)
void eq_vq_argmax_kernel(const float* __restrict__ x,     // N x 64
                         const float* __restrict__ cb,    // V x 64
                         const float* __restrict__ xrn,   // N
                         const float* __restrict__ cbrn,  // V
                         float* __restrict__ zq,          // N x 64
                         int* __restrict__ idx_out,       // N
                         int V) {
    const int lane = threadIdx.x;      // 0..31
    const int l16  = lane & 15;
    const int half = lane >> 4;        // 0 | 1
    const int row0 = blockIdx.x * 16;

    // ---- A tile: x_norm rows row0..row0+15, all 16 K-chunks resident.
    // f32 A 16x4 layout: lane = M; VGPR0/1 hold K{0,1} (lanes 0-15), K{2,3} (16-31).
    const int am = row0 + l16;
    const float arn = xrn[am];
    const float* arow = x + (size_t)am * DTOK + 2 * half;
    v2f a[16];
#pragma unroll
    for (int kc = 0; kc < 16; ++kc) {
        const float2 t = *(const float2*)(arow + 4 * kc);
        a[kc].x = t.x * arn;
        a[kc].y = t.y * arn;
    }

    float vmax[8];
    int   vidx[8];
#pragma unroll
    for (int r = 0; r < 8; ++r) { vmax[r] = NEG_INF; vidx[r] = 0; }

    const int nchunks = V >> 4;        // 256 for V=4096
    for (int cc = 0; cc < nchunks; ++cc) {
        const int bn = (cc << 4) + l16;          // this lane's column (code id)
        const float brn = cbrn[bn];
        const float* brow = cb + (size_t)bn * DTOK + 2 * half;
        if (cc + 1 < nchunks) {                   // global_prefetch_b8 next chunk
            __builtin_prefetch(cb + (size_t)(bn + 16) * DTOK, 0, 0);
        }

        v8f c = {};                               // fresh f32 accumulator
#pragma unroll
        for (int kc = 0; kc < 16; ++kc) {
            const float2 t = *(const float2*)(brow + 4 * kc);
            v2f b;
            b.x = t.x * brn;
            b.y = t.y * brn;
            // D = A(16x4) * B(4x16) + C, full-precision f32
            c = __builtin_amdgcn_wmma_f32_16x16x4_f32(
                    /*neg_a=*/false, a[kc], /*neg_b=*/false, b,
                    /*c_mod=*/(short)0, c, /*reuse_a=*/false, /*reuse_b=*/false);
        }

        // C layout: VGPR r -> row (r + 8*half), lane%16 -> column bn.
#pragma unroll
        for (int r = 0; r < 8; ++r) {
            const float s = c[r];
            if (s > vmax[r]) { vmax[r] = s; vidx[r] = bn; }
        }
    }

    // ---- argmax butterfly within each 16-lane half (tie -> lowest index).
#pragma unroll
    for (int r = 0; r < 8; ++r) {
        float m = vmax[r];
        int   id = vidx[r];
#pragma unroll
        for (int s = 8; s >= 1; s >>= 1) {
            const float om = __shfl_xor(m, s, 16);
            const int   oi = __shfl_xor(id, s, 16);
            if (om > m || (om == m && oi < id)) { m = om; id = oi; }
        }
        vmax[r] = m;
        vidx[r] = id;
    }

    // Lane 0 holds rows 0..7 (r=row), lane 16 holds rows 8..15 (r=row-8).
    __shared__ int sidx[16];
    if (l16 == 0) {
#pragma unroll
        for (int r = 0; r < 8; ++r) {
            const int row = r + 8 * half;
            sidx[row] = vidx[r];
            idx_out[row0 + row] = vidx[r];
        }
    }
    __syncthreads();

    // ---- gather un-normalized codebook rows -> z_q (forward value).
    // Each lane copies 32 floats: row = lane%16, column half = lane/16.
    const int grow = l16;
    const int gidx = sidx[grow];
    const float4* sp = (const float4*)(cb + (size_t)gidx * DTOK + half * 32);
    float4* dp = (float4*)(zq + (size_t)(row0 + grow) * DTOK + half * 32);
#pragma unroll
    for (int i = 0; i < 8; ++i) dp[i] = sp[i];
}

// ---------------------------------------------------------------------------
extern "C" void kernel_launch(void* const* d_in, const int* in_sizes, int n_in,
                              void* d_out, int out_size, void* d_ws, size_t ws_size,
                              hipStream_t stream) {
    const float* x  = (const float*)d_in[0];   // (B,T,D) f32
    const float* cb = (const float*)d_in[1];   // (V,D)   f32
    // d_in[2] = gumbel_noise: cancels in the forward value -> never read.

    const int N = in_sizes[0] / DTOK;          // 16384
    const int V = in_sizes[1] / DTOK;          // 4096

    float* xrn  = (float*)d_ws;                // N floats
    float* cbrn = xrn + N;                     // V floats  (total 80 KB scratch)

    float* zq  = (float*)d_out;                // N*64 floats
    int*   idx = (int*)(zq + (size_t)N * DTOK); // N int32, bit-cast tail

    eq_rnorm_kernel<<<(N + 255) / 256, 256, 0, stream>>>(x, xrn, N);
    eq_rnorm_kernel<<<(V + 255) / 256, 256, 0, stream>>>(cb, cbrn, V);
    eq_vq_argmax_kernel<<<N / 16, 32, 0, stream>>>(x, cb, xrn, cbrn, zq, idx, V);
}